// Metalayer_51711406243970
// MI455X (gfx1250) — compile-verified
//
#include <hip/hip_runtime.h>

// ---------------------------------------------------------------------------
// MI455X (gfx1250, wave32) implementation.
// Heavy path: 1024x1024 fp32 GEMMs on V_WMMA_F32_16X16X4_F32, staged with
// CDNA5 async global->LDS DMA (GLOBAL_LOAD_ASYNC_TO_LDS_B128, ASYNCcnt),
// double-buffered LDS tiles (DMA(t+1) overlaps WMMA compute(t)).
// Occupancy: 64x128 block tiles (4 waves) -> 128 blocks/problem, and the
// launcher batches two independent GEMMs per launch (256 blocks) for the
// Taylor/squaring phases of expm.
//   - Newton-Schulz inverse of Cmat (pure GEMM)
//   - EigM = Cinv @ (B.Cmat + Kmat), pre-scaled by WH*2^-20
//   - expm via Taylor(8) + 20 squarings, complex arithmetic as real GEMM pairs
// Working set (~50MB) resides in the 192MB L2; compute-bound on matrix pipes.
// ---------------------------------------------------------------------------

typedef __attribute__((ext_vector_type(2))) float v2f;
typedef __attribute__((ext_vector_type(4))) float v4f;
typedef __attribute__((ext_vector_type(8))) float v8f;

#define MT   1024
#define TOTN 66176
#define KB   16     // K-tile depth
#define LDSA 20     // A tile LDS row stride (floats): 16+4 pad, 16B-aligned rows
#define LDSB 128    // B tile LDS row stride (floats): natural [k][n]

// ---- LDS offset of a generic pointer that points into shared memory ----
__device__ __forceinline__ unsigned lds_off(const void* p) {
    return (unsigned)(size_t)(const __attribute__((address_space(3))) char*)p;
}

// ---- async DMA: 16 bytes per lane, global -> LDS, tracked by ASYNCcnt ----
__device__ __forceinline__ void async_copy_b128(unsigned ldst, const float* g) {
    asm volatile("global_load_async_to_lds_b128 %0, %1, off"
                 :: "v"(ldst), "v"((unsigned long long)(size_t)g)
                 : "memory");
}

__device__ __forceinline__ void wait_async0() {
#if __has_builtin(__builtin_amdgcn_s_wait_asynccnt)
    __builtin_amdgcn_s_wait_asynccnt(0);
#else
    asm volatile("s_wait_asynccnt 0" ::: "memory");
#endif
}

// ---------------------------------------------------------------------------
// fp32 WMMA GEMM:  Out = alpha*A@B + beta*Cin + gamma*I      (all 1024x1024)
// Block = 128 threads = 4 waves (1x4 along N); block tile 64x128;
// wave tile 64x32 = 4x2 WMMA accumulators. Grid: 128 blocks per problem;
// two independent problems batched via blockIdx.x >> 7.
// ---------------------------------------------------------------------------
__global__ __launch_bounds__(128)
void gemm_f32_wmma(const float* A0, const float* B0, const float* C0, float* O0,
                   float al0, float be0, float ga0,
                   const float* A1, const float* B1, const float* C1, float* O1,
                   float al1, float be1, float ga1)
{
    __shared__ float As[2][64 * LDSA];    // A tile [m][k], padded stride
    __shared__ float Bs[2][KB * LDSB];    // B tile natural [k][n]

    const int prob = blockIdx.x >> 7;
    const float* __restrict__ A   = prob ? A1 : A0;
    const float* __restrict__ B   = prob ? B1 : B0;
    const float* __restrict__ Cin = prob ? C1 : C0;
    float* __restrict__       Out = prob ? O1 : O0;
    const float alpha = prob ? al1 : al0;
    const float beta  = prob ? be1 : be0;
    const float gamma = prob ? ga1 : ga0;

    const int blk   = blockIdx.x & 127;
    const int tid   = threadIdx.x;
    const int lane  = tid & 31;
    const int wave  = tid >> 5;
    const int bm    = (blk >> 3) * 64;    // 16 M-tiles
    const int bn    = (blk & 7) * 128;    // 8 N-tiles
    const int wn    = wave * 32;          // 0,32,64,96
    const int l15   = lane & 15;
    const int khalf = (lane >> 4) * 2;    // lanes 0-15 -> K{0,1}; 16-31 -> K{2,3}

    // Issue one K-tile (A 64xKB, B KBx128) as 6 async b128 ops per thread.
    auto issue_tile = [&](int buf, int k0) {
#pragma unroll
        for (int it = 0; it < 2; ++it) {
            int idx = tid + it * 128;            // 0..255 A float4 slots
            int r   = idx >> 2;                  // A row 0..63
            int c4  = idx & 3;                   // A float4-col 0..3
            async_copy_b128(lds_off(&As[buf][r * LDSA + c4 * 4]),
                            A + (size_t)(bm + r) * MT + k0 + c4 * 4);
        }
#pragma unroll
        for (int it = 0; it < 4; ++it) {
            int idx = tid + it * 128;            // 0..511 B float4 slots
            int kk  = idx >> 5;                  // B k-row 0..15
            int n4  = idx & 31;                  // B float4-col 0..31
            async_copy_b128(lds_off(&Bs[buf][kk * LDSB + n4 * 4]),
                            B + (size_t)(k0 + kk) * MT + bn + n4 * 4);
        }
    };

    v8f acc[4][2];
#pragma unroll
    for (int i = 0; i < 4; ++i)
#pragma unroll
        for (int j = 0; j < 2; ++j)
            acc[i][j] = {};

    issue_tile(0, 0);
    for (int t = 0; t < MT / KB; ++t) {
        const int buf = t & 1;
        wait_async0();          // this wave's tile-t DMA complete (in-order)
        __syncthreads();        // => everyone's tile-t data visible; buf^1 free
        if (t + 1 < MT / KB)
            issue_tile(buf ^ 1, (t + 1) * KB);   // DMA overlaps compute below

        const float* Ac = As[buf];
        const float* Bc = Bs[buf];
#pragma unroll
        for (int kk = 0; kk < KB; kk += 4) {
            v2f af[4], bf[2];
#pragma unroll
            for (int i = 0; i < 4; ++i)
                af[i] = *(const v2f*)(&Ac[(i * 16 + l15) * LDSA + kk + khalf]);
#pragma unroll
            for (int j = 0; j < 2; ++j) {
                const float* bp = &Bc[(kk + khalf) * LDSB + wn + j * 16 + l15];
                v2f b; b[0] = bp[0]; b[1] = bp[LDSB];   // ds_load_2addr_b32
                bf[j] = b;
            }
#pragma unroll
            for (int i = 0; i < 4; ++i)
#pragma unroll
                for (int j = 0; j < 2; ++j)
                    acc[i][j] = __builtin_amdgcn_wmma_f32_16x16x4_f32(
                        false, af[i], false, bf[j], (short)0, acc[i][j], false, false);
        }
    }

    // ---- epilogue: Out = alpha*acc + beta*Cin + gamma*I ----
    const int hsel = lane >> 4;  // VGPR r holds M = r + 8*hsel
#pragma unroll
    for (int i = 0; i < 4; ++i) {
#pragma unroll
        for (int j = 0; j < 2; ++j) {
            int col = bn + wn + j * 16 + l15;
#pragma unroll
            for (int r = 0; r < 8; ++r) {
                int row = bm + i * 16 + r + 8 * hsel;
                float v = alpha * acc[i][j][r];
                if (beta != 0.0f)  v += beta * Cin[(size_t)row * MT + col];
                if (gamma != 0.0f && row == col) v += gamma;
                Out[(size_t)row * MT + col] = v;
            }
        }
    }
}

// ---------------------------------------------------------------------------
// Small kernels (<<1% of FLOPs)
// ---------------------------------------------------------------------------
__global__ void fill_kernel(float* p, int n) {
    int i = blockIdx.x * blockDim.x + threadIdx.x;
    if (i < n) p[i] = 0.0f;
}

// sigmoid -> hs; neffnn MLP [1,64,64,2]; eta; Bdiag
__global__ void prep_kernel(const float* __restrict__ hp,
                            const float* nW0, const float* nW1, const float* nW2,
                            const float* nb0, const float* nb1, const float* nb2,
                            float* hs, float* neffs, float* eta, float* Bd)
{
    int n = blockIdx.x * blockDim.x + threadIdx.x;
    if (n >= 512) return;
    float h = 1.0f / (1.0f + expf(-hp[n]));
    h = h * 0.2f + 0.1f;
    hs[n] = h;
    float h1[64], h2[64];
    for (int j = 0; j < 64; ++j) h1[j] = fmaxf(h * nW0[j] + nb0[j], 0.0f);
    for (int j = 0; j < 64; ++j) {
        float s = nb1[j];
        for (int k = 0; k < 64; ++k) s += h1[k] * nW1[k * 64 + j];
        h2[j] = fmaxf(s, 0.0f);
    }
    for (int m = 0; m < 2; ++m) {
        float s = nb2[m];
        for (int k = 0; k < 64; ++k) s += h2[k] * nW2[k * 2 + m];
        neffs[n * 2 + m] = s;
        eta[n * 2 + m]   = s / (s + 1.0f);
        Bd[n * 2 + m]    = s * 11.8119f;
    }
}

// enn MLP [1,256,256,1536], one block per waveguide
__global__ __launch_bounds__(256)
void enn_kernel(const float* __restrict__ hs,
                const float* W0, const float* b0, const float* W1, const float* b1,
                const float* W2, const float* b2, float* __restrict__ Ey)
{
    __shared__ float h1[256], h2[256];
    int n = blockIdx.x, j = threadIdx.x;
    float h = hs[n];
    h1[j] = fmaxf(h * W0[j] + b0[j], 0.0f);
    __syncthreads();
    float s = b1[j];
    for (int k = 0; k < 256; ++k) s += h1[k] * W1[k * 256 + j];
    h2[j] = fmaxf(s, 0.0f);
    __syncthreads();
    for (int o = 0; o < 6; ++o) {
        int f = j + o * 256;  // 0..1535 = m*768+fy
        float t = b2[f];
        for (int k = 0; k < 256; ++k) t += h2[k] * W2[k * 1536 + f];
        Ey[(size_t)n * 1536 + f] = t;
    }
}

// U0[n*2+m] = 0.02 * eta * dot(Ey[n,m,:], E0[n*128 : n*128+768])
__global__ __launch_bounds__(256)
void u0_kernel(const float* __restrict__ Ey, const float* __restrict__ E0,
               const float* __restrict__ eta, float* __restrict__ U0)
{
    __shared__ float red[256];
    int nm = blockIdx.x, n = nm >> 1, m = nm & 1;
    float s = 0.0f;
    for (int f = threadIdx.x; f < 768; f += 256)
        s += Ey[(size_t)n * 1536 + m * 768 + f] * E0[n * 128 + f];
    red[threadIdx.x] = s; __syncthreads();
    for (int w = 128; w; w >>= 1) {
        if (threadIdx.x < w) red[threadIdx.x] += red[threadIdx.x + w];
        __syncthreads();
    }
    if (threadIdx.x == 0) U0[nm] = 0.02f * eta[nm] * red[0];
}

// gen_C / gen_K MLPs [3,128,128,128,4], one block per (n, delta) pair
__global__ __launch_bounds__(128)
void ck_kernel(const float* __restrict__ hs,
               const float* cW0, const float* cb0, const float* cW1, const float* cb1,
               const float* cW2, const float* cb2, const float* cW3, const float* cb3,
               const float* kW0, const float* kb0, const float* kW1, const float* kb1,
               const float* kW2, const float* kb2, const float* kW3, const float* kb3,
               float* __restrict__ Cvals, float* __restrict__ Kvals)
{
    __shared__ float buf[2][128];
    int r = blockIdx.x, n = r / 6, d = r % 6 - 2;
    int jn = n + d; jn = jn < 0 ? 0 : (jn > 511 ? 511 : jn);
    float hi = hs[n], hj = hs[jn], dis = (float)d * 0.4f;
    int j = threadIdx.x;

    // ---- C net ----
    buf[0][j] = fmaxf(hi * cW0[j] + hj * cW0[128 + j] + dis * cW0[256 + j] + cb0[j], 0.0f);
    __syncthreads();
    { float s = cb1[j]; for (int k = 0; k < 128; ++k) s += buf[0][k] * cW1[k * 128 + j];
      buf[1][j] = fmaxf(s, 0.0f); } __syncthreads();
    { float s = cb2[j]; for (int k = 0; k < 128; ++k) s += buf[1][k] * cW2[k * 128 + j];
      buf[0][j] = fmaxf(s, 0.0f); } __syncthreads();
    if (j < 4) { float t = cb3[j]; for (int k = 0; k < 128; ++k) t += buf[0][k] * cW3[k * 4 + j];
                 Cvals[r * 4 + j] = t; }
    __syncthreads();

    // ---- K net ----
    buf[0][j] = fmaxf(hi * kW0[j] + hj * kW0[128 + j] + dis * kW0[256 + j] + kb0[j], 0.0f);
    __syncthreads();
    { float s = kb1[j]; for (int k = 0; k < 128; ++k) s += buf[0][k] * kW1[k * 128 + j];
      buf[1][j] = fmaxf(s, 0.0f); } __syncthreads();
    { float s = kb2[j]; for (int k = 0; k < 128; ++k) s += buf[1][k] * kW2[k * 128 + j];
      buf[0][j] = fmaxf(s, 0.0f); } __syncthreads();
    if (j < 4) { float t = kb3[j]; for (int k = 0; k < 128; ++k) t += buf[0][k] * kW3[k * 4 + j];
                 Kvals[r * 4 + j] = t; }
}

// banded COO scatter-add into dense [1024,1024] (duplicates at clipped edges)
__global__ void scatter_kernel(const float* __restrict__ Cvals, const float* __restrict__ Kvals,
                               float* __restrict__ Cmat, float* __restrict__ Kmat)
{
    int e = blockIdx.x * blockDim.x + threadIdx.x;
    if (e >= 3072 * 4) return;
    int rc = e >> 2, c = e & 3;
    int n = rc / 6, d = rc % 6 - 2;
    int jn = n + d; jn = jn < 0 ? 0 : (jn > 511 ? 511 : jn);
    int R  = n * 2 + (c >> 1);
    int Cc = jn * 2 + (c & 1);
    atomicAdd(&Cmat[(size_t)R * MT + Cc], Cvals[e]);
    atomicAdd(&Kmat[(size_t)R * MT + Cc], Kvals[e]);
}

// W = Bdiag[row] * Cmat + Kmat
__global__ void form_w_kernel(const float* __restrict__ Cmat, const float* __restrict__ Kmat,
                              const float* __restrict__ Bd, float* __restrict__ W)
{
    int i = blockIdx.x * blockDim.x + threadIdx.x;
    if (i < MT * MT) W[i] = Bd[i >> 10] * Cmat[i] + Kmat[i];
}

__global__ __launch_bounds__(256)
void rowsum_kernel(const float* __restrict__ A, float* __restrict__ rs)
{
    __shared__ float red[256];
    int row = blockIdx.x;
    float s = 0.0f;
    for (int j = threadIdx.x; j < MT; j += 256) s += fabsf(A[(size_t)row * MT + j]);
    red[threadIdx.x] = s; __syncthreads();
    for (int w = 128; w; w >>= 1) {
        if (threadIdx.x < w) red[threadIdx.x] += red[threadIdx.x + w];
        __syncthreads();
    }
    if (threadIdx.x == 0) rs[row] = red[0];
}

__global__ void colsum_kernel(const float* __restrict__ A, float* __restrict__ cs)
{
    int c = blockIdx.x * blockDim.x + threadIdx.x;
    if (c >= MT) return;
    float s = 0.0f;
    for (int i = 0; i < MT; ++i) s += fabsf(A[(size_t)i * MT + c]);
    cs[c] = s;
}

__global__ __launch_bounds__(256)
void max_scale_kernel(const float* __restrict__ rs, const float* __restrict__ cs,
                      float* __restrict__ scl)
{
    __shared__ float ra[256], rb[256];
    float a = 0.0f, b = 0.0f;
    for (int i = threadIdx.x; i < MT; i += 256) { a = fmaxf(a, rs[i]); b = fmaxf(b, cs[i]); }
    ra[threadIdx.x] = a; rb[threadIdx.x] = b; __syncthreads();
    for (int w = 128; w; w >>= 1) {
        if (threadIdx.x < w) {
            ra[threadIdx.x] = fmaxf(ra[threadIdx.x], ra[threadIdx.x + w]);
            rb[threadIdx.x] = fmaxf(rb[threadIdx.x], rb[threadIdx.x + w]);
        }
        __syncthreads();
    }
    if (threadIdx.x == 0) scl[0] = 1.0f / (ra[0] * rb[0]);
}

// X0 = A^T * (1/(||A||_1 ||A||_inf))  — Newton-Schulz seed
__global__ void transpose_scale_kernel(const float* __restrict__ A, const float* __restrict__ scl,
                                       float* __restrict__ X)
{
    int i = blockIdx.x * blockDim.x + threadIdx.x;
    if (i >= MT * MT) return;
    int r = i >> 10, c = i & 1023;
    X[i] = A[(size_t)c * MT + r] * scl[0];
}

__global__ void init_complex_kernel(float* __restrict__ Tr, float* __restrict__ Ti)
{
    int i = blockIdx.x * blockDim.x + threadIdx.x;
    if (i >= MT * MT) return;
    int r = i >> 10, c = i & 1023;
    Tr[i] = (r == c) ? 1.0f : 0.0f;
    Ti[i] = 0.0f;
}

__global__ __launch_bounds__(256)
void matvec_kernel(const float* __restrict__ P, const float* __restrict__ x,
                   float* __restrict__ y)
{
    __shared__ float red[256];
    int row = blockIdx.x;
    float s = 0.0f;
    for (int j = threadIdx.x; j < MT; j += 256) s += P[(size_t)row * MT + j] * x[j];
    red[threadIdx.x] = s; __syncthreads();
    for (int w = 128; w; w >>= 1) {
        if (threadIdx.x < w) red[threadIdx.x] += red[threadIdx.x + w];
        __syncthreads();
    }
    if (threadIdx.x == 0) y[row] = red[0];
}

// gather form of the overlapping scatter-add: En[t] = sum over <=6 windows
__global__ void en_kernel(const float* __restrict__ Ey, const float* __restrict__ eta,
                          const float* __restrict__ Uzr, const float* __restrict__ Uzi,
                          float* __restrict__ out)
{
    int t = blockIdx.x * blockDim.x + threadIdx.x;
    if (t >= TOTN) return;
    int n_lo = (t >= 768) ? ((t - 640) >> 7) : 0;
    int n_hi = t >> 7; if (n_hi > 511) n_hi = 511;
    float ar = 0.0f, ai = 0.0f;
    for (int n = n_lo; n <= n_hi; ++n) {
        int f = t - (n << 7);
        float e0 = eta[n * 2 + 0], e1 = eta[n * 2 + 1];
        float y0 = Ey[(size_t)n * 1536 + f];
        float y1 = Ey[(size_t)n * 1536 + 768 + f];
        ar += e0 * y0 * Uzr[n * 2 + 0] + e1 * y1 * Uzr[n * 2 + 1];
        ai += e0 * y0 * Uzi[n * 2 + 0] + e1 * y1 * Uzi[n * 2 + 1];
    }
    out[t] = ar;
    out[TOTN + t] = ai;
}

// ---------------------------------------------------------------------------
// Host orchestration
// ---------------------------------------------------------------------------
extern "C" void kernel_launch(void* const* d_in, const int* in_sizes, int n_in,
                              void* d_out, int out_size, void* d_ws, size_t ws_size,
                              hipStream_t stream)
{
    const float* E0 = (const float*)d_in[0];
    const float* hp = (const float*)d_in[1];
    const float *nW0,*nW1,*nW2,*nb0,*nb1,*nb2;
    const float *cW0,*cW1,*cW2,*cW3,*cb0,*cb1,*cb2,*cb3;
    const float *kW0,*kW1,*kW2,*kW3,*kb0,*kb1,*kb2,*kb3;
    const float *eW0,*eW1,*eW2,*eb0,*eb1,*eb2;

    if (n_in >= 30) {  // pytree leaves flattened individually
        nW0=(const float*)d_in[2];  nW1=(const float*)d_in[3];  nW2=(const float*)d_in[4];
        nb0=(const float*)d_in[5];  nb1=(const float*)d_in[6];  nb2=(const float*)d_in[7];
        cW0=(const float*)d_in[8];  cW1=(const float*)d_in[9];  cW2=(const float*)d_in[10]; cW3=(const float*)d_in[11];
        cb0=(const float*)d_in[12]; cb1=(const float*)d_in[13]; cb2=(const float*)d_in[14]; cb3=(const float*)d_in[15];
        kW0=(const float*)d_in[16]; kW1=(const float*)d_in[17]; kW2=(const float*)d_in[18]; kW3=(const float*)d_in[19];
        kb0=(const float*)d_in[20]; kb1=(const float*)d_in[21]; kb2=(const float*)d_in[22]; kb3=(const float*)d_in[23];
        eW0=(const float*)d_in[24]; eW1=(const float*)d_in[25]; eW2=(const float*)d_in[26];
        eb0=(const float*)d_in[27]; eb1=(const float*)d_in[28]; eb2=(const float*)d_in[29];
    } else {           // each dict value concatenated flat
        const float* p;
        p=(const float*)d_in[2]; nW0=p; nW1=p+64;  nW2=p+64+4096;
        p=(const float*)d_in[3]; nb0=p; nb1=p+64;  nb2=p+128;
        p=(const float*)d_in[4]; cW0=p; cW1=p+384; cW2=p+384+16384; cW3=p+384+32768;
        p=(const float*)d_in[5]; cb0=p; cb1=p+128; cb2=p+256; cb3=p+384;
        p=(const float*)d_in[6]; kW0=p; kW1=p+384; kW2=p+384+16384; kW3=p+384+32768;
        p=(const float*)d_in[7]; kb0=p; kb1=p+128; kb2=p+256; kb3=p+384;
        p=(const float*)d_in[8]; eW0=p; eW1=p+256; eW2=p+256+65536;
        p=(const float*)d_in[9]; eb0=p; eb1=p+256; eb2=p+512;
    }

    float* ws = (float*)d_ws;
    float* hs    = ws + 0;
    float* neffs = ws + 512;
    float* eta   = ws + 1536;
    float* Bd    = ws + 2560;
    float* U0    = ws + 3584;
    float* Uzr   = ws + 4608;
    float* Uzi   = ws + 5632;
    float* rs    = ws + 6656;
    float* cs    = ws + 7680;
    float* scl   = ws + 8704;
    float* Cvals = ws + 16384;
    float* Kvals = ws + 32768;
    float* Ey    = ws + 65536;            // 512*1536 floats
    const size_t MB = (size_t)1 << 20;    // 1M floats per matrix slot
    float* Cmat = ws + 1*MB;  float* Kmat = ws + 2*MB;  float* Wm = ws + 3*MB;
    float* X    = ws + 4*MB;  float* X2   = ws + 5*MB;  float* Tt = ws + 6*MB;
    float* S0   = ws + 7*MB;  float* Tr   = ws + 8*MB;  float* Ti = ws + 9*MB;
    float* Ur   = ws + 10*MB; float* Ui   = ws + 11*MB;

    // single GEMM: 128 blocks; dual independent GEMMs: 256 blocks
    auto GEMM = [&](const float* A, const float* B, const float* C, float* O,
                    float al, float be, float ga) {
        gemm_f32_wmma<<<128, 128, 0, stream>>>(A, B, C, O, al, be, ga,
                                               A, B, C, O, al, be, ga);
    };
    auto GEMM2 = [&](const float* A0, const float* B0, const float* C0, float* O0,
                     float al0, float be0, float ga0,
                     const float* A1, const float* B1, const float* C1, float* O1,
                     float al1, float be1, float ga1) {
        gemm_f32_wmma<<<256, 128, 0, stream>>>(A0, B0, C0, O0, al0, be0, ga0,
                                               A1, B1, C1, O1, al1, be1, ga1);
    };

    // --- small networks + banded matrices ---
    prep_kernel<<<2, 256, 0, stream>>>(hp, nW0, nW1, nW2, nb0, nb1, nb2, hs, neffs, eta, Bd);
    enn_kernel<<<512, 256, 0, stream>>>(hs, eW0, eb0, eW1, eb1, eW2, eb2, Ey);
    u0_kernel<<<1024, 256, 0, stream>>>(Ey, E0, eta, U0);
    ck_kernel<<<3072, 128, 0, stream>>>(hs, cW0, cb0, cW1, cb1, cW2, cb2, cW3, cb3,
                                        kW0, kb0, kW1, kb1, kW2, kb2, kW3, kb3, Cvals, Kvals);
    fill_kernel<<<(2 * 1048576 + 255) / 256, 256, 0, stream>>>(Cmat, 2 * 1048576); // Cmat+Kmat
    scatter_kernel<<<(12288 + 255) / 256, 256, 0, stream>>>(Cvals, Kvals, Cmat, Kmat);
    form_w_kernel<<<4096, 256, 0, stream>>>(Cmat, Kmat, Bd, Wm);

    // --- Newton-Schulz inverse of Cmat (pure GEMM, dependent chain) ---
    rowsum_kernel<<<1024, 256, 0, stream>>>(Cmat, rs);
    colsum_kernel<<<4, 256, 0, stream>>>(Cmat, cs);
    max_scale_kernel<<<1, 256, 0, stream>>>(rs, cs, scl);
    transpose_scale_kernel<<<4096, 256, 0, stream>>>(Cmat, scl, X);
    for (int it = 0; it < 30; ++it) {
        GEMM(Cmat, X, X, Tt, 1.0f, 0.0f, 0.0f);    // T  = C @ X
        GEMM(X, Tt, X, X2, -1.0f, 2.0f, 0.0f);     // X' = 2X - X@T
        float* sw = X; X = X2; X2 = sw;
    }

    // --- EigM pre-scaled: S0 = (WH * 2^-20) * Cinv @ W ---
    GEMM(X, Wm, X, S0, 0.75f * (1.0f / 1048576.0f), 0.0f, 0.0f);

    // --- expm(i*S): Taylor order 8 on A = i*S0; (Tr',Ti') are independent ---
    init_complex_kernel<<<4096, 256, 0, stream>>>(Tr, Ti);
    for (int k = 8; k >= 1; --k) {
        float inv = 1.0f / (float)k;
        GEMM2(S0, Ti, S0, Ur, -inv, 0.0f, 1.0f,    // Tr' = I - (S0@Ti)/k
              S0, Tr, S0, Ui,  inv, 0.0f, 0.0f);   // Ti' = (S0@Tr)/k
        float* sw = Tr; Tr = Ur; Ur = sw; sw = Ti; Ti = Ui; Ui = sw;
    }
    // --- 20 squarings: (Tr+iTi)^2, two independent pairs per squaring ---
    for (int s = 0; s < 20; ++s) {
        GEMM2(Tr, Tr, Tr, Ur, 1.0f, 0.0f, 0.0f,    // Ur  = Tr@Tr
              Tr, Ti, Tr, Ui, 1.0f, 0.0f, 0.0f);   // Ui  = Tr@Ti
        GEMM2(Ti, Ti, Ur, Ur, -1.0f, 1.0f, 0.0f,   // Ur -= Ti@Ti
              Ti, Tr, Ui, Ui,  1.0f, 1.0f, 0.0f);  // Ui += Ti@Tr
        float* sw = Tr; Tr = Ur; Ur = sw; sw = Ti; Ti = Ui; Ui = sw;
    }

    // --- Uz = P @ U0, then overlapping-window field assembly ---
    matvec_kernel<<<1024, 256, 0, stream>>>(Tr, U0, Uzr);
    matvec_kernel<<<1024, 256, 0, stream>>>(Ti, U0, Uzi);
    en_kernel<<<(TOTN + 255) / 256, 256, 0, stream>>>(Ey, eta, Uzr, Uzi, (float*)d_out);
}